// GridSample_22007412425229
// MI455X (gfx1250) — compile-verified
//
#include <hip/hip_runtime.h>
#include <hip/hip_bf16.h>
#include <stdint.h>

// ---------------- problem constants (from reference setup_inputs) ------------
#define B_   2
#define N_   6
#define NCAM (B_ * N_)      // 12
#define C_   64
#define FH   16
#define FW   44
#define CHW  (FH * FW)      // 704 floats per channel plane
#define GD   10
#define GH   128
#define GW   128
#define PTS_PER_CAM (GD * GH * GW)   // 163840

// ---------------- tiling --------------------------------------------------
#define CPH 32                        // channels per block (half of C)
#define SMEM_FLOATS (CPH * CHW)       // 22528 floats = 88 KB LDS
#define PPT 8                         // points per thread
#define THREADS 256
#define PTS_PER_BLOCK (THREADS * PPT) // 2048
#define CHUNKS (PTS_PER_CAM / PTS_PER_BLOCK) // 80
#define NBLOCKS (NCAM * 2 * CHUNKS)   // 1920

typedef float v4f __attribute__((ext_vector_type(4)));

// =====================================================================
// Main kernel: async global->LDS staging (ASYNCcnt), LDS gathers,
// non-temporal b128 output stores (kernel is HBM-store-bound:
// 503 MB / 23.3 TB/s ~= 21.6 us floor).
// =====================================================================
__global__ __launch_bounds__(THREADS) void bev_grid_sample(
    const float* __restrict__ feat,   // (B,N,C,FH,FW)
    const float* __restrict__ prm,
    float* __restrict__ out)          // (B,N,C,GD,GH,GW)
{
  __shared__ float smem[SMEM_FLOATS];   // 88 KB: 32 channel planes of 16x44

  const int blk   = blockIdx.x;
  const int chunk = blk % CHUNKS;
  const int half  = (blk / CHUNKS) & 1;
  const int cam   = blk / (CHUNKS * 2);
  const unsigned tid = threadIdx.x;

  // ---- Stage this block's 32-channel feature slab into LDS via the CDNA5
  // ---- async global->LDS copy path (ASYNCcnt-tracked).
  {
    const float* src = feat + (size_t)cam * (C_ * CHW) + (size_t)half * SMEM_FLOATS;
    unsigned long long gbase = (unsigned long long)src;
    // Low 32 bits of a generic LDS pointer are the wave-relative LDS byte
    // offset (LDS aperture is 4GB-aligned). Also captures smem for aliasing.
    unsigned ldsBase = (unsigned)(uintptr_t)(void*)smem;
#pragma unroll
    for (int k = 0; k < (SMEM_FLOATS / 4) / THREADS; ++k) {   // 22 x b128 per thread
      unsigned byteOff = (tid + (unsigned)k * THREADS) * 16u;
      unsigned dstOff  = ldsBase + byteOff;
      asm volatile("global_load_async_to_lds_b128 %0, %1, %2"
                   :
                   : "v"(dstOff), "v"(byteOff), "s"(gbase)
                   : "memory");
    }
#if __has_builtin(__builtin_amdgcn_s_wait_asynccnt)
    __builtin_amdgcn_s_wait_asynccnt(0);
#else
    asm volatile("s_wait_asynccnt 0" ::: "memory");
#endif
    // Make the asm's writes to smem visible to the optimizer: without this,
    // smem has no visible stores and all its loads get folded to undef.
    asm volatile("" : : "v"((void*)smem) : "memory");
  }
  __syncthreads();

  // ---- Per-camera folded projection params (uniform across block).
  const float* Pp = prm + cam * 18;
  const float A00 = Pp[0], A01 = Pp[1], A02 = Pp[2];
  const float A10 = Pp[3], A11 = Pp[4], A12 = Pp[5];
  const float A20 = Pp[6], A21 = Pp[7], A22 = Pp[8];
  const float b0 = Pp[9], b1 = Pp[10], b2 = Pp[11];
  const float P00 = Pp[12], P01 = Pp[13], P10 = Pp[14], P11 = Pp[15];
  const float q0 = Pp[16], q1 = Pp[17];
  const float sx = prm[216], sy = prm[217];

  // ---- Compute bilinear taps for PPT consecutive points.
  const int p0 = chunk * PTS_PER_BLOCK + (int)tid * PPT;
  int   off[PPT][4];
  float wgt[PPT][4];
#pragma unroll
  for (int j = 0; j < PPT; ++j) {
    const int p = p0 + j;
    const int w = p & (GW - 1);
    const int h = (p >> 7) & (GH - 1);
    const int d = p >> 14;
    const float x = -50.8f + 0.8f * (float)w;
    const float y = -50.8f + 0.8f * (float)h;
    const float z = -4.6f + 0.8f * (float)d;

    const float ci0 = A00 * x + A01 * y + A02 * z + b0;
    const float ci1 = A10 * x + A11 * y + A12 * z + b1;
    const float ci2 = A20 * x + A21 * y + A22 * z + b2;
    float u = ci0 / ci2;
    float v = ci1 / ci2;
    if (v < 0.0f) { u = -10000.0f; v = -10000.0f; }       // kept_not on v, both chans
    if (fabsf(u) > 10000.0f) u = -1000.0f;
    if (fabsf(v) > 10000.0f) v = -1000.0f;

    const float cx = P00 * u + P01 * v + q0;
    const float cy = P10 * u + P11 * v + q1;
    const float ix = cx * sx;                              // align_corners fold
    const float iy = cy * sy;

    const float fx0 = floorf(ix), fy0 = floorf(iy);
    const float fx1 = fx0 + 1.0f, fy1 = fy0 + 1.0f;
    const float wx1 = ix - fx0, wx0 = 1.0f - wx1;
    const float wy1 = iy - fy0, wy0 = 1.0f - wy1;
    const bool vx0 = (fx0 >= 0.0f) && (fx0 <= (float)(FW - 1));
    const bool vx1 = (fx1 >= 0.0f) && (fx1 <= (float)(FW - 1));
    const bool vy0 = (fy0 >= 0.0f) && (fy0 <= (float)(FH - 1));
    const bool vy1 = (fy1 >= 0.0f) && (fy1 <= (float)(FH - 1));
    const int x0 = (int)fminf(fmaxf(fx0, 0.0f), (float)(FW - 1));
    const int x1 = (int)fminf(fmaxf(fx1, 0.0f), (float)(FW - 1));
    const int y0 = (int)fminf(fmaxf(fy0, 0.0f), (float)(FH - 1));
    const int y1 = (int)fminf(fmaxf(fy1, 0.0f), (float)(FH - 1));
    off[j][0] = y0 * FW + x0;
    off[j][1] = y0 * FW + x1;
    off[j][2] = y1 * FW + x0;
    off[j][3] = y1 * FW + x1;
    wgt[j][0] = (vx0 && vy0) ? wx0 * wy0 : 0.0f;
    wgt[j][1] = (vx1 && vy0) ? wx1 * wy0 : 0.0f;
    wgt[j][2] = (vx0 && vy1) ? wx0 * wy1 : 0.0f;
    wgt[j][3] = (vx1 && vy1) ? wx1 * wy1 : 0.0f;
  }

  // ---- 32 channels: gather 4 taps/point from LDS, stream out as NT b128s.
  float* outBase = out + ((size_t)cam * C_ + (size_t)half * CPH) * PTS_PER_CAM + p0;
#pragma unroll 2
  for (int c = 0; c < CPH; ++c) {
    const float* fc = smem + c * CHW;
    float acc[PPT];
#pragma unroll
    for (int j = 0; j < PPT; ++j) {
      acc[j] = wgt[j][0] * fc[off[j][0]]
             + wgt[j][1] * fc[off[j][1]]
             + wgt[j][2] * fc[off[j][2]]
             + wgt[j][3] * fc[off[j][3]];
    }
    float* dst = outBase + (size_t)c * PTS_PER_CAM;
    v4f lo = {acc[0], acc[1], acc[2], acc[3]};
    v4f hi = {acc[4], acc[5], acc[6], acc[7]};
    __builtin_nontemporal_store(lo, (v4f*)dst);
    __builtin_nontemporal_store(hi, (v4f*)(dst + 4));
  }
}

// =====================================================================
// Tiny per-camera parameter folding (12 threads, negligible cost).
//   [cam*18 + 0..8]  A = intrins * inv(rots) * bda
//   [cam*18 + 9..11] bv = -intrins*inv(rots)*trans
//   [cam*18 +12..15] P = post_rots[:2,:2]
//   [cam*18 +16..17] q = post_trans[:2]
//   [216] sx = (FW-1)/(imgW-1),  [217] sy = (FH-1)/(imgH-1)
// =====================================================================
__global__ void bev_precompute(const float* __restrict__ rots,
                               const float* __restrict__ trans,
                               const float* __restrict__ intrins,
                               const float* __restrict__ post_rots,
                               const float* __restrict__ post_trans,
                               const float* __restrict__ bda,
                               const int* __restrict__ imgH,
                               const int* __restrict__ imgW,
                               float* __restrict__ prm) {
  int cam = threadIdx.x;
  if (cam == 0) {
    prm[216] = (float)(FW - 1) / (float)(imgW[0] - 1);
    prm[217] = (float)(FH - 1) / (float)(imgH[0] - 1);
  }
  if (cam >= NCAM) return;
  int b = cam / N_;

  const float* R = rots + cam * 9;
  float r00 = R[0], r01 = R[1], r02 = R[2];
  float r10 = R[3], r11 = R[4], r12 = R[5];
  float r20 = R[6], r21 = R[7], r22 = R[8];
  float det = r00 * (r11 * r22 - r12 * r21)
            - r01 * (r10 * r22 - r12 * r20)
            + r02 * (r10 * r21 - r11 * r20);
  float id = 1.0f / det;
  float Ri[3][3] = {
      {(r11 * r22 - r12 * r21) * id, (r02 * r21 - r01 * r22) * id, (r01 * r12 - r02 * r11) * id},
      {(r12 * r20 - r10 * r22) * id, (r00 * r22 - r02 * r20) * id, (r02 * r10 - r00 * r12) * id},
      {(r10 * r21 - r11 * r20) * id, (r01 * r20 - r00 * r21) * id, (r00 * r11 - r01 * r10) * id}};

  const float* K = intrins + cam * 9;
  float M[3][3];  // M = K * Rinv
  for (int i = 0; i < 3; ++i)
    for (int j = 0; j < 3; ++j)
      M[i][j] = K[i * 3 + 0] * Ri[0][j] + K[i * 3 + 1] * Ri[1][j] + K[i * 3 + 2] * Ri[2][j];

  const float* G = bda + b * 9;
  const float* t = trans + cam * 3;
  float* P = prm + cam * 18;
  for (int i = 0; i < 3; ++i) {
    for (int j = 0; j < 3; ++j)
      P[i * 3 + j] = M[i][0] * G[0 * 3 + j] + M[i][1] * G[1 * 3 + j] + M[i][2] * G[2 * 3 + j];
    P[9 + i] = -(M[i][0] * t[0] + M[i][1] * t[1] + M[i][2] * t[2]);
  }
  const float* pr = post_rots + cam * 9;
  P[12] = pr[0]; P[13] = pr[1]; P[14] = pr[3]; P[15] = pr[4];
  P[16] = post_trans[cam * 3 + 0];
  P[17] = post_trans[cam * 3 + 1];
}

extern "C" void kernel_launch(void* const* d_in, const int* in_sizes, int n_in,
                              void* d_out, int out_size, void* d_ws, size_t ws_size,
                              hipStream_t stream) {
  const float* features   = (const float*)d_in[0];
  const float* rots       = (const float*)d_in[1];
  const float* trans      = (const float*)d_in[2];
  const float* intrins    = (const float*)d_in[3];
  const float* post_rots  = (const float*)d_in[4];
  const float* post_trans = (const float*)d_in[5];
  const float* bda        = (const float*)d_in[6];
  const int*   imgH       = (const int*)d_in[7];
  const int*   imgW       = (const int*)d_in[8];
  float* out = (float*)d_out;
  float* prm = (float*)d_ws;

  bev_precompute<<<1, 32, 0, stream>>>(rots, trans, intrins, post_rots,
                                       post_trans, bda, imgH, imgW, prm);
  bev_grid_sample<<<NBLOCKS, THREADS, 0, stream>>>(features, prm, out);
}